// GATClassifier_78915729097083
// MI455X (gfx1250) — compile-verified
//
#include <hip/hip_runtime.h>
#include <math.h>

// ---------------------------------------------------------------------------
// GATClassifier for gfx1250 (MI455X).  fp32 WMMA for x@W, atomics for the
// segment softmax / aggregation, fused BN+ELU, pooled FC head.
// ---------------------------------------------------------------------------

typedef __attribute__((ext_vector_type(2))) float v2f;
typedef __attribute__((ext_vector_type(8))) float v8f;

#define HIDDEN    128
#define HEADS     4
#define OUT_CH    32
#define NEG_SLOPE 0.2f
#define BN_EPS    1e-5f

// ---------------------------------------------------------------------------
// h = x @ W   (x: [n,128], W: [128,128] row-major, h: [n,128])
// One block = 256 threads = 8 waves. Block handles one 16-row tile; wave w
// handles column tile w (16 cols). K-loop: 32 x v_wmma_f32_16x16x4_f32.
// EXEC must be all-ones for WMMA: ragged rows handled by clamped loads and a
// single uniform-branch store fallback, not per-element predication.
// ---------------------------------------------------------------------------
__global__ void __launch_bounds__(256)
gat_gemm_xw(const float* __restrict__ X, const float* __restrict__ W,
            float* __restrict__ H, int n)
{
    const int wave = threadIdx.x >> 5;      // 0..7 -> column tile
    const int lane = threadIdx.x & 31;
    const int row0 = blockIdx.x * 16;
    const int col0 = wave * 16;
    const int half = lane >> 4;             // 0: lanes 0-15, 1: lanes 16-31
    const int mn   = lane & 15;             // m (A rows) / n (B cols)

    int rowA = row0 + mn;
    if (rowA >= n) rowA = n - 1;            // clamp (keep EXEC all-ones)
    const float* __restrict__ xrow = X + (size_t)rowA * HIDDEN + 2 * half;
    const float* __restrict__ wcol = W + (size_t)(2 * half) * HIDDEN + col0 + mn;

    v8f acc = {};
    for (int k0 = 0; k0 < HIDDEN; k0 += 4) {
        // A 16x4: lane half 0 holds K = k0+0,k0+1 ; half 1 holds K = k0+2,k0+3
        v2f a, b;
        a.x = xrow[k0 + 0];
        a.y = xrow[k0 + 1];
        // B 4x16: vgpr0 = K (k0 + 2*half), vgpr1 = K+1, lane%16 = column
        b.x = wcol[(size_t)k0 * HIDDEN];
        b.y = wcol[(size_t)k0 * HIDDEN + HIDDEN];
        acc = __builtin_amdgcn_wmma_f32_16x16x4_f32(
            /*neg_a=*/false, a, /*neg_b=*/false, b,
            /*c_mod=*/(short)0, acc, /*reuse_a=*/false, /*reuse_b=*/false);
    }

    // C/D layout: VGPR v -> M = v + 8*half, N = mn
    float* __restrict__ hout = H + (size_t)(row0 + 8 * half) * HIDDEN + col0 + mn;
    if (row0 + 16 <= n) {                   // uniform fast path: full tile
        #pragma unroll
        for (int v = 0; v < 8; ++v)
            hout[(size_t)v * HIDDEN] = acc[v];
    } else {                                // cold ragged tail
        #pragma unroll
        for (int v = 0; v < 8; ++v) {
            int r = row0 + v + 8 * half;
            if (r < n) H[(size_t)r * HIDDEN + col0 + mn] = acc[v];
        }
    }
}

// ---------------------------------------------------------------------------
// Per (node, head): alpha_src / alpha_dst dot products; init segment max to
// -inf and denom to 0 for this layer.
// ---------------------------------------------------------------------------
__global__ void gat_alpha_prep(const float* __restrict__ H,
                               const float* __restrict__ att_src,
                               const float* __restrict__ att_dst,
                               float* __restrict__ as_, float* __restrict__ ad_,
                               float* __restrict__ m_, float* __restrict__ den_,
                               int n)
{
    int idx = blockIdx.x * blockDim.x + threadIdx.x;   // node*HEADS + head
    if (idx >= n * HEADS) return;
    int head = idx & (HEADS - 1);
    int node = idx >> 2;
    const float4* h4 = (const float4*)(H + (size_t)node * HIDDEN + head * OUT_CH);
    const float4* s4 = (const float4*)(att_src + head * OUT_CH);
    const float4* d4 = (const float4*)(att_dst + head * OUT_CH);
    float s = 0.f, d = 0.f;
    #pragma unroll
    for (int j = 0; j < OUT_CH / 4; ++j) {
        float4 h = h4[j], a = s4[j], b = d4[j];
        s += h.x * a.x + h.y * a.y + h.z * a.z + h.w * a.w;
        d += h.x * b.x + h.y * b.y + h.z * b.z + h.w * b.w;
    }
    as_[idx] = s;
    ad_[idx] = d;
    m_[idx]  = -INFINITY;
    den_[idx] = 0.f;
}

__global__ void zero_f32(float* __restrict__ p, size_t cnt)
{
    size_t i = (size_t)blockIdx.x * blockDim.x + threadIdx.x;
    if (i < cnt) p[i] = 0.f;
}

// float atomic max via ordered-bits int/uint atomics (works with -inf init)
__device__ __forceinline__ void atomicMaxF(float* addr, float val)
{
    if (val >= 0.f) atomicMax((int*)addr, __float_as_int(val));
    else            atomicMin((unsigned int*)addr, __float_as_uint(val));
}

__device__ __forceinline__ void edge_src_dst(const int* __restrict__ ei,
                                             int e, int E, int& s, int& d)
{
    if (e < E) { s = ei[e]; d = ei[E + e]; }      // edge_index[0][e], [1][e]
    else       { s = d = e - E; }                 // appended self-loop
}

__global__ void gat_edge_max(const int* __restrict__ ei, int E, int n,
                             const float* __restrict__ as_,
                             const float* __restrict__ ad_,
                             float* __restrict__ m_)
{
    int idx = blockIdx.x * blockDim.x + threadIdx.x;   // (edge)*HEADS + head
    int tot = (E + n) * HEADS;
    if (idx >= tot) return;
    int head = idx & (HEADS - 1);
    int e = idx >> 2;
    int s, d; edge_src_dst(ei, e, E, s, d);
    float sc = as_[s * HEADS + head] + ad_[d * HEADS + head];
    sc = sc > 0.f ? sc : sc * NEG_SLOPE;               // leaky_relu
    atomicMaxF(&m_[d * HEADS + head], sc);
}

__global__ void gat_edge_exp(const int* __restrict__ ei, int E, int n,
                             const float* __restrict__ as_,
                             const float* __restrict__ ad_,
                             const float* __restrict__ m_,
                             float* __restrict__ den_,
                             float* __restrict__ ew)
{
    int idx = blockIdx.x * blockDim.x + threadIdx.x;
    int tot = (E + n) * HEADS;
    if (idx >= tot) return;
    int head = idx & (HEADS - 1);
    int e = idx >> 2;
    int s, d; edge_src_dst(ei, e, E, s, d);
    float sc = as_[s * HEADS + head] + ad_[d * HEADS + head];
    sc = sc > 0.f ? sc : sc * NEG_SLOPE;
    float ex = __expf(sc - m_[d * HEADS + head]);
    ew[idx] = ex;
    atomicAdd(&den_[d * HEADS + head], ex);
}

// One wave per edge: 32 lanes x float4 = 128 channels; scatter h[src]*alpha.
__global__ void __launch_bounds__(256)
gat_edge_msg(const int* __restrict__ ei, int E, int n,
             const float* __restrict__ H,
             const float* __restrict__ den_,
             const float* __restrict__ ew,
             float* __restrict__ acc)
{
    int wid  = (int)(((size_t)blockIdx.x * blockDim.x + threadIdx.x) >> 5);
    int lane = threadIdx.x & 31;
    if (wid >= E + n) return;
    int s, d; edge_src_dst(ei, wid, E, s, d);
    int c0   = lane * 4;                 // channel group, stays inside a head
    int head = c0 >> 5;
    float alpha = ew[(size_t)wid * HEADS + head] / den_[d * HEADS + head];
    float4 hv = *(const float4*)(H + (size_t)s * HIDDEN + c0);
    float* out = acc + (size_t)d * HIDDEN + c0;
    atomicAdd(out + 0, hv.x * alpha);
    atomicAdd(out + 1, hv.y * alpha);
    atomicAdd(out + 2, hv.z * alpha);
    atomicAdd(out + 3, hv.w * alpha);
}

// bias + batchnorm (running stats) + ELU, elementwise (float4 per thread)
__global__ void gat_post(const float4* __restrict__ acc,
                         const float* __restrict__ bias,
                         const float* __restrict__ gamma,
                         const float* __restrict__ beta,
                         const float* __restrict__ mean,
                         const float* __restrict__ var,
                         float4* __restrict__ xout, size_t total4)
{
    size_t idx = (size_t)blockIdx.x * blockDim.x + threadIdx.x;
    if (idx >= total4) return;
    int c = (int)((idx * 4) & (HIDDEN - 1));
    float4 v = acc[idx];
    float4 r;
    {
        float t = v.x + bias[c + 0];
        t = (t - mean[c + 0]) * (gamma[c + 0] * rsqrtf(var[c + 0] + BN_EPS)) + beta[c + 0];
        r.x = t > 0.f ? t : __expf(t) - 1.f;
        t = v.y + bias[c + 1];
        t = (t - mean[c + 1]) * (gamma[c + 1] * rsqrtf(var[c + 1] + BN_EPS)) + beta[c + 1];
        r.y = t > 0.f ? t : __expf(t) - 1.f;
        t = v.z + bias[c + 2];
        t = (t - mean[c + 2]) * (gamma[c + 2] * rsqrtf(var[c + 2] + BN_EPS)) + beta[c + 2];
        r.z = t > 0.f ? t : __expf(t) - 1.f;
        t = v.w + bias[c + 3];
        t = (t - mean[c + 3]) * (gamma[c + 3] * rsqrtf(var[c + 3] + BN_EPS)) + beta[c + 3];
        r.w = t > 0.f ? t : __expf(t) - 1.f;
    }
    xout[idx] = r;
}

__global__ void gat_pool(const float* __restrict__ x,
                         const int* __restrict__ batch,
                         float* __restrict__ sums, float* __restrict__ cnt,
                         int n)
{
    size_t idx = (size_t)blockIdx.x * blockDim.x + threadIdx.x;
    if (idx >= (size_t)n * HIDDEN) return;
    int node = (int)(idx >> 7);
    int c    = (int)(idx & (HIDDEN - 1));
    int b = batch[node];
    atomicAdd(&sums[(size_t)b * HIDDEN + c], x[idx]);
    if (c == 0) atomicAdd(&cnt[b], 1.f);
}

// One block (64 threads) per graph: mean, fc1+relu, fc2.
__global__ void __launch_bounds__(64)
gat_head(const float* __restrict__ sums, const float* __restrict__ cnt,
         const float* __restrict__ fc1w, const float* __restrict__ fc1b,
         const float* __restrict__ fc2w, const float* __restrict__ fc2b,
         float* __restrict__ logits)
{
    __shared__ float p[HIDDEN];
    __shared__ float hm[HIDDEN / 2];
    int g = blockIdx.x, t = threadIdx.x;
    float inv = 1.f / fmaxf(cnt[g], 1.f);
    p[t]      = sums[(size_t)g * HIDDEN + t] * inv;
    p[t + 64] = sums[(size_t)g * HIDDEN + t + 64] * inv;
    __syncthreads();
    float a = fc1b[t];
    #pragma unroll 4
    for (int k = 0; k < HIDDEN; ++k) a += p[k] * fc1w[k * 64 + t];
    hm[t] = fmaxf(a, 0.f);
    __syncthreads();
    if (t < 10) {
        float o = fc2b[t];
        #pragma unroll 4
        for (int k = 0; k < 64; ++k) o += hm[k] * fc2w[k * 10 + t];
        logits[g * 10 + t] = o;
    }
}

// ---------------------------------------------------------------------------
static inline int cdiv(long long a, long long b) { return (int)((a + b - 1) / b); }

extern "C" void kernel_launch(void* const* d_in, const int* in_sizes, int n_in,
                              void* d_out, int out_size, void* d_ws, size_t ws_size,
                              hipStream_t stream)
{
    const float* x0      = (const float*)d_in[0];
    const int*   ei      = (const int*)  d_in[1];   // [2, E]
    const int*   batch   = (const int*)  d_in[2];
    const float* W       = (const float*)d_in[3];   // [3,128,128]
    const float* att_src = (const float*)d_in[4];   // [3,4,32]
    const float* att_dst = (const float*)d_in[5];
    const float* bias    = (const float*)d_in[6];   // [3,128]
    const float* gamma   = (const float*)d_in[7];
    const float* beta    = (const float*)d_in[8];
    const float* mean    = (const float*)d_in[9];
    const float* var     = (const float*)d_in[10];
    const float* fc1w    = (const float*)d_in[11];  // [128,64]
    const float* fc1b    = (const float*)d_in[12];
    const float* fc2w    = (const float*)d_in[13];  // [64,10]
    const float* fc2b    = (const float*)d_in[14];

    const int n    = in_sizes[0] / HIDDEN;   // 50000
    const int E    = in_sizes[1] / 2;        // 800000
    const int Etot = E + n;                  // + self loops
    const size_t nf = (size_t)n * HIDDEN;

    // Workspace carve-up (fp32 elements)
    float* h    = (float*)d_ws;              // [n,128]
    float* acc  = h    + nf;                 // [n,128]
    float* xbuf = acc  + nf;                 // [n,128]
    float* as_  = xbuf + nf;                 // [n,4]
    float* ad_  = as_  + (size_t)n * HEADS;
    float* m_   = ad_  + (size_t)n * HEADS;
    float* den_ = m_   + (size_t)n * HEADS;
    float* ew   = den_ + (size_t)n * HEADS;  // [Etot,4]
    float* sums = ew   + (size_t)Etot * HEADS; // [256,128]
    float* cnt  = sums + 256 * HIDDEN;         // [256] (contiguous with sums)

    const float* xin = x0;
    for (int l = 0; l < 3; ++l) {
        gat_gemm_xw<<<cdiv(n, 16), 256, 0, stream>>>(
            xin, W + (size_t)l * HIDDEN * HIDDEN, h, n);
        gat_alpha_prep<<<cdiv((long long)n * HEADS, 256), 256, 0, stream>>>(
            h, att_src + l * HEADS * OUT_CH, att_dst + l * HEADS * OUT_CH,
            as_, ad_, m_, den_, n);
        zero_f32<<<cdiv((long long)nf, 256), 256, 0, stream>>>(acc, nf);
        gat_edge_max<<<cdiv((long long)Etot * HEADS, 256), 256, 0, stream>>>(
            ei, E, n, as_, ad_, m_);
        gat_edge_exp<<<cdiv((long long)Etot * HEADS, 256), 256, 0, stream>>>(
            ei, E, n, as_, ad_, m_, den_, ew);
        gat_edge_msg<<<cdiv(Etot, 8), 256, 0, stream>>>(
            ei, E, n, h, den_, ew, acc);
        gat_post<<<cdiv((long long)(nf / 4), 256), 256, 0, stream>>>(
            (const float4*)acc, bias + l * HIDDEN, gamma + l * HIDDEN,
            beta + l * HIDDEN, mean + l * HIDDEN, var + l * HIDDEN,
            (float4*)xbuf, nf / 4);
        xin = xbuf;
    }

    zero_f32<<<cdiv(256 * HIDDEN + 256, 256), 256, 0, stream>>>(
        sums, (size_t)256 * HIDDEN + 256);
    gat_pool<<<cdiv((long long)nf, 256), 256, 0, stream>>>(xbuf, batch, sums, cnt, n);
    gat_head<<<256, 64, 0, stream>>>(sums, cnt, fc1w, fc1b, fc2w, fc2b,
                                     (float*)d_out);
}